// CorrelationLoss_5274219840259
// MI455X (gfx1250) — compile-verified
//
#include <hip/hip_runtime.h>

namespace {
constexpr int       kRows  = 192;                         // N*C
constexpr int       kHW    = 512 * 512;                   // 262144, power of two
constexpr long long kTotal = (long long)kRows * kHW;      // 50331648 pairs/elements
constexpr int       kBPR   = 16;                          // blocks per row
constexpr int       kChunk = kHW / kBPR;                  // 16384 elements per block
constexpr int       kTPB   = 256;                         // 8 waves (wave32)
constexpr int       kF4    = kChunk / 4 / kTPB;           // 16 float4 per thread
constexpr int       kBins  = 64;                          // 8x8 histogram
constexpr float     kEps   = 1e-10f;
}

typedef float v2f __attribute__((ext_vector_type(2)));
typedef float v8f __attribute__((ext_vector_type(8)));

__global__ __launch_bounds__(64) void hist_init_kernel(unsigned* __restrict__ hist) {
  hist[threadIdx.x] = 0u;
}

__device__ __forceinline__ int bin_of(float v) {
  int b = (int)(v * 8.0f);            // floor for v >= 0
  b = b < 0 ? 0 : (b > 7 ? 7 : b);    // clip like jnp.clip(...,0,7)
  return b;
}

__global__ __launch_bounds__(kTPB) void stream_reduce_kernel(
    const float* __restrict__ x,
    float* __restrict__ partials,          // [gridDim.x][3]
    unsigned* __restrict__ ghist) {        // [64]
  __shared__ unsigned lhist[8 * kBins];    // per-wave private histograms
  __shared__ float    wsum[8][3];

  const int tid  = threadIdx.x;
  const int wave = tid >> 5;
  const int lane = tid & 31;

  lhist[tid]       = 0u;
  lhist[tid + 256] = 0u;
  __syncthreads();

  const int row = blockIdx.x / kBPR;
  const int chk = blockIdx.x - row * kBPR;
  const long long base = (long long)row * kHW + (long long)chk * kChunk;
  const bool last_chunk = (chk == kBPR - 1);

  // even/odd accumulators per quantity -> 64 partials per wave (WMMA B operand)
  float a1 = 0.f, b1 = 0.f;   // S1
  float a2 = 0.f, b2 = 0.f;   // S2
  float a3 = 0.f, b3 = 0.f;   // Sxy

#pragma unroll
  for (int k = 0; k < kF4; ++k) {
    const int fi = tid + k * kTPB;                  // coalesced float4 index in chunk
    const long long off = base + 4LL * fi;
    const float4 v = *(const float4*)(x + off);

    float ncor, nhist;
    if (last_chunk && fi == (kChunk / 4 - 1)) {     // last float4 of this row
      const long long noff = off + 4;
      nhist = (noff < kTotal) ? x[noff] : x[0];     // global roll wrap
      ncor  = x[(long long)row * kHW];              // per-row circular wrap
    } else {
      const float nx = x[off + 4];                  // adjacent element: cache hit
      ncor  = nx;
      nhist = nx;
    }

    const float s1 = (v.x + v.y) + (v.z + v.w);
    const float s2 = v.x * v.x + v.y * v.y + v.z * v.z + v.w * v.w;
    const float s3 = v.x * v.y + v.y * v.z + v.z * v.w + v.w * ncor;
    if (k & 1) { b1 += s1; b2 += s2; b3 += s3; }
    else       { a1 += s1; a2 += s2; a3 += s3; }

    const int bx = bin_of(v.x), by = bin_of(v.y), bz = bin_of(v.z),
              bw = bin_of(v.w), bn = bin_of(nhist);
    unsigned* h = lhist + wave * kBins;
    atomicAdd(&h[bx * 8 + by], 1u);
    atomicAdd(&h[by * 8 + bz], 1u);
    atomicAdd(&h[bz * 8 + bw], 1u);
    atomicAdd(&h[bw * 8 + bn], 1u);
  }

  // ---- intra-wave reduction via V_WMMA_F32_16X16X4_F32 ----
  // A = ones(16x4), B holds the 64 per-wave partials: D[m][n] = sum_k B[k][n].
  // Summing D[0][n] over the 16 columns (shfl_xor 1/2/4/8) yields the full
  // 64-partial wave sum in f32. EXEC is all 1s here (no divergence).
  v2f ones; ones.x = 1.0f; ones.y = 1.0f;
  v8f c0 = {};
  v2f p;
  p.x = a1; p.y = b1;
  v8f d1 = __builtin_amdgcn_wmma_f32_16x16x4_f32(false, ones, false, p, (short)0, c0, false, false);
  p.x = a2; p.y = b2;
  v8f d2 = __builtin_amdgcn_wmma_f32_16x16x4_f32(false, ones, false, p, (short)0, c0, false, false);
  p.x = a3; p.y = b3;
  v8f d3 = __builtin_amdgcn_wmma_f32_16x16x4_f32(false, ones, false, p, (short)0, c0, false, false);
  float r1 = d1[0], r2 = d2[0], r3 = d3[0];
#pragma unroll
  for (int m = 1; m <= 8; m <<= 1) {   // masks stay within each 16-lane half
    r1 += __shfl_xor(r1, m, 32);
    r2 += __shfl_xor(r2, m, 32);
    r3 += __shfl_xor(r3, m, 32);
  }
  if (lane == 0) { wsum[wave][0] = r1; wsum[wave][1] = r2; wsum[wave][2] = r3; }
  __syncthreads();

  if (tid < kBins) {                    // flush per-wave histograms
    unsigned s = 0;
#pragma unroll
    for (int w = 0; w < 8; ++w) s += lhist[w * kBins + tid];
    if (s) atomicAdd(&ghist[tid], s);
  }
  if (tid == 0) {                       // fixed-order cross-wave combine
    float s1 = 0.f, s2 = 0.f, s3 = 0.f;
#pragma unroll
    for (int w = 0; w < 8; ++w) { s1 += wsum[w][0]; s2 += wsum[w][1]; s3 += wsum[w][2]; }
    float* pp = partials + (long long)blockIdx.x * 3;
    pp[0] = s1; pp[1] = s2; pp[2] = s3;
  }
}

__global__ __launch_bounds__(256) void finalize_kernel(
    const float* __restrict__ partials,
    const unsigned* __restrict__ ghist,
    float* __restrict__ out) {
  __shared__ float sc[kRows];
  __shared__ float sh[kBins];
  const int t = threadIdx.x;

  if (t < kRows) {
    float s1 = 0.f, s2 = 0.f, s3 = 0.f;
    for (int c = 0; c < kBPR; ++c) {           // fixed order -> deterministic
      const float* pp = partials + (t * kBPR + c) * 3;
      s1 += pp[0]; s2 += pp[1]; s3 += pp[2];
    }
    const float inv = 1.0f / (float)kHW;
    const float m   = s1 * inv;
    const float var = s2 * inv - m * m;        // var_x == var_y
    const float cov = s3 * inv - m * m;        // mean of rolled row == mean
    sc[t] = fabsf(cov / (var + kEps));
  }
  if (t < kBins) {
    const float h = (float)ghist[t] / (float)kTotal;
    const float d = h - 1.0f / (float)kBins;
    sh[t] = d * d;
  }
  __syncthreads();
  if (t == 0) {
    float cl = 0.f;
    for (int i = 0; i < kRows; ++i) cl += sc[i];
    float hl = 0.f;
    for (int i = 0; i < kBins; ++i) hl += sh[i];
    out[0] = cl / (float)kRows + hl / (float)kBins;   // LAMBDA_COR = LAMBDA_2D = 1
  }
}

extern "C" void kernel_launch(void* const* d_in, const int* in_sizes, int n_in,
                              void* d_out, int out_size, void* d_ws, size_t ws_size,
                              hipStream_t stream) {
  (void)in_sizes; (void)n_in; (void)out_size; (void)ws_size;
  const float* x = (const float*)d_in[0];

  float*    partials = (float*)d_ws;                                   // 3072*3 f32
  unsigned* ghist    = (unsigned*)((char*)d_ws +
                        (size_t)kRows * kBPR * 3 * sizeof(float));     // 64 u32

  hist_init_kernel<<<1, 64, 0, stream>>>(ghist);
  stream_reduce_kernel<<<kRows * kBPR, kTPB, 0, stream>>>(x, partials, ghist);
  finalize_kernel<<<1, 256, 0, stream>>>(partials, ghist, (float*)d_out);
}